// CorrBlock_22557168238853
// MI455X (gfx1250) — compile-verified
//
#include <hip/hip_runtime.h>
#include <hip/hip_bf16.h>

typedef float v2f __attribute__((ext_vector_type(2)));
typedef float v8f __attribute__((ext_vector_type(8)));
typedef int   v4i_b __attribute__((vector_size(16)));   // matches builtin param

#define HAS_ASYNC_LDS (__has_builtin(__builtin_amdgcn_global_load_async_to_lds_b128) && \
                       __has_builtin(__builtin_amdgcn_s_wait_asynccnt))

namespace {
constexpr int kCin     = 320;
constexpr int kCout    = 160;
constexpr int kT       = 9;
constexpr int kHW      = 1024;          // 32*32
constexpr int kNB      = 32;            // b*(T-1) = 4*8
constexpr int kCStride = kT * kHW;      // channel stride in x: 9216 floats
}

// ---------------------------------------------------------------------------
// Kernel 1: f[j][n][c][i] = sum_k w_j[c][k] * x[b][k][t+j][i]
// grid.x = 64 (j,n), grid.y = 10 (row tiles of 16), block = 128 (4 waves)
// each wave sweeps 16 column tiles; per tile: 80 x wmma_f32_16x16x4_f32
// ---------------------------------------------------------------------------
__global__ __launch_bounds__(128)
void proj_kernel(const float* __restrict__ x,
                 const float* __restrict__ w0,
                 const float* __restrict__ w1,
                 float* __restrict__ f) {
  const int idx = blockIdx.x;
  const int j = idx >> 5;
  const int n = idx & 31;
  const int b = n >> 3;
  const int t = n & 7;
  const int frame = t + j;                 // j==0 -> frames 0..7, j==1 -> 1..8
  const float* __restrict__ wgt = j ? w1 : w0;

  const int mt   = blockIdx.y << 4;        // output-channel tile base (0..144)
  const int lane = threadIdx.x & 31;
  const int wv   = threadIdx.x >> 5;
  const int lr   = lane & 15;              // row within fragment
  const int hi   = lane >> 4;              // k-pair / row-half selector

  const float* __restrict__ xb =
      x + ((size_t)b * kCin * kT + frame) * kHW;
  float* __restrict__ dst = f + ((size_t)j * kNB + n) * kCout * kHW;

  for (int nt4 = wv; nt4 < 64; nt4 += 4) {
    const int nt = nt4 << 4;               // column tile base
    v8f acc = {};
    for (int k0 = 0; k0 < kCin; k0 += 4) {
      const int ka = k0 + (hi << 1);
      v2f a, bf;
      a.x = wgt[(mt + lr) * kCin + ka];
      a.y = wgt[(mt + lr) * kCin + ka + 1];
      bf.x = xb[(size_t)ka * kCStride + nt + lr];
      bf.y = xb[(size_t)(ka + 1) * kCStride + nt + lr];
      acc = __builtin_amdgcn_wmma_f32_16x16x4_f32(
          false, a, false, bf, (short)0, acc, false, false);
    }
#pragma unroll
    for (int r = 0; r < 8; ++r)
      dst[(size_t)(mt + r + (hi << 3)) * kHW + nt + lr] = acc[r];
  }
}

// ---------------------------------------------------------------------------
// Kernel 2 (fused): per block (n, mb) compute 32 corr rows (one image row of
// pixels) entirely in LDS, then pool + bilinear-sample both pyramid levels.
//   corr[n][i][j] = (1/160) * sum_c f0[n][c][i] * f1[n][c][j],  i = mb*32 + c
// A tile (f0 columns) staged into LDS via async global->LDS; B streamed from
// L2; each wave runs dual accumulators (2 row-tiles) per column tile so every
// B fragment feeds two WMMAs. corr never touches HBM.
// ---------------------------------------------------------------------------
__global__ __launch_bounds__(256)
void corr_fused_kernel(const float* __restrict__ f, float* __restrict__ out) {
  constexpr int ROWP  = 1025;              // 32x32 image row pitch (+1 pad)
  constexpr int PROWP = 257;               // 16x16 pooled pitch (+1 pad)
  __shared__ float As[kCout * 32];         // As[k][m] : A^T tile, 20 KB
  __shared__ float corrS[32 * ROWP];       // 32 corr rows, ~128 KB
  __shared__ float poolS[32 * PROWP];      // 32 pooled images, ~32 KB

  const int mb   = blockIdx.x;             // 0..31 == cy of this block
  const int n    = blockIdx.y;             // 0..31
  const int i0   = mb << 5;
  const int tid  = threadIdx.x;
  const int lane = tid & 31;
  const int wv   = tid >> 5;
  const int lr   = lane & 15;
  const int hi   = lane >> 4;
  const int b    = n >> 3;
  const int t    = n & 7;

  const float* __restrict__ f0 = f + (size_t)n * kCout * kHW;
  const float* __restrict__ f1 = f + (size_t)(kNB + n) * kCout * kHW;

  // ---- stage As[k][0..31] = f0[k][i0 .. i0+31] (5120 floats) ----
#if HAS_ASYNC_LDS
  {
    typedef __attribute__((address_space(1))) v4i_b* g4p;   // global int4*
    typedef __attribute__((address_space(3))) v4i_b* l4p;   // LDS int4*
#pragma unroll
    for (int p = 0; p < 5; ++p) {
      const int e = (p * 256 + tid) * 4;   // float index, 16B chunks
      const int k = e >> 5;
      const int c = e & 31;
      __builtin_amdgcn_global_load_async_to_lds_b128(
          (g4p)(f0 + (size_t)k * kHW + i0 + c),
          (l4p)(&As[k * 32 + c]),
          0, 0);
    }
    __builtin_amdgcn_s_wait_asynccnt(0);
  }
#else
  for (int e = tid; e < kCout * 32; e += 256) {
    const int k = e >> 5;
    const int c = e & 31;
    As[k * 32 + c] = f0[(size_t)k * kHW + i0 + c];
  }
#endif
  __syncthreads();

  // ---- GEMM: 32 x 1024, K=160, results scaled into LDS ----
  const float scale = 1.0f / (float)kCout;
  for (int ct = wv; ct < 64; ct += 8) {
    const int nt = ct << 4;
    v8f acc0 = {};
    v8f acc1 = {};
    for (int k0 = 0; k0 < kCout; k0 += 4) {
      const int ka = k0 + (hi << 1);
      v2f bfr, a0, a1;
      bfr.x = f1[(size_t)ka * kHW + nt + lr];
      bfr.y = f1[(size_t)(ka + 1) * kHW + nt + lr];
      a0.x = As[ka * 32 + lr];
      a0.y = As[(ka + 1) * 32 + lr];
      a1.x = As[ka * 32 + 16 + lr];
      a1.y = As[(ka + 1) * 32 + 16 + lr];
      acc0 = __builtin_amdgcn_wmma_f32_16x16x4_f32(
          false, a0, false, bfr, (short)0, acc0, false, false);
      acc1 = __builtin_amdgcn_wmma_f32_16x16x4_f32(
          false, a1, false, bfr, (short)0, acc1, false, false);
    }
#pragma unroll
    for (int r = 0; r < 8; ++r) {
      const int row = r + (hi << 3);
      corrS[row * ROWP + nt + lr]        = acc0[r] * scale;
      corrS[(row + 16) * ROWP + nt + lr] = acc1[r] * scale;
    }
  }
  __syncthreads();

  // ---- 2x2 average pool: 32 images of 32x32 -> 16x16 ----
  for (int o = tid; o < 32 * 256; o += 256) {
    const int m  = o >> 8;
    const int y  = (o >> 4) & 15;
    const int x2 = o & 15;
    const float* src = &corrS[m * ROWP + (y * 2) * 32 + x2 * 2];
    poolS[m * PROWP + y * 16 + x2] =
        0.25f * (src[0] + src[1] + src[32] + src[33]);
  }
  __syncthreads();

  // ---- window sampling: 98 channels x 32 pixels, straight from LDS ----
  // out[b][lvl*49 + a*7 + bb][t][cy][cx]; dx = a-3 on x, dy = bb-3 on y.
  const int c  = tid & 31;                 // pixel cx; its image = corrS row c
  const int kq = tid >> 5;
  const int cy = mb;
  for (int k = kq; k < 98; k += 8) {
    float val;
    if (k < 49) {
      const int a  = k / 7;
      const int bb = k - a * 7;
      const int sx = c + a - 3;
      const int sy = cy + bb - 3;
      const bool inb = (sx >= 0) & (sx < 32) & (sy >= 0) & (sy < 32);
      val = inb ? corrS[c * ROWP + sy * 32 + sx] : 0.0f;
    } else {
      const int kk = k - 49;
      const int a  = kk / 7;
      const int bb = kk - a * 7;
      const float fx = c * 0.5f + (float)(a - 3);
      const float fy = cy * 0.5f + (float)(bb - 3);
      const float x0f = floorf(fx), y0f = floorf(fy);
      const float wx = fx - x0f, wy = fy - y0f;
      const int x0 = (int)x0f, y0 = (int)y0f;
      const float* img = &poolS[c * PROWP];
      auto corner = [&](int yy, int xx) -> float {
        const bool v = (xx >= 0) & (xx < 16) & (yy >= 0) & (yy < 16);
        return v ? img[yy * 16 + xx] : 0.0f;
      };
      val = corner(y0, x0)         * (1.0f - wx) * (1.0f - wy)
          + corner(y0, x0 + 1)     * wx          * (1.0f - wy)
          + corner(y0 + 1, x0)     * (1.0f - wx) * wy
          + corner(y0 + 1, x0 + 1) * wx          * wy;
    }
    out[(((size_t)b * 98 + k) * 8 + t) * kHW + (cy << 5) + c] = val;
  }
}

// ---------------------------------------------------------------------------
extern "C" void kernel_launch(void* const* d_in, const int* in_sizes, int n_in,
                              void* d_out, int out_size, void* d_ws, size_t ws_size,
                              hipStream_t stream) {
  const float* x  = (const float*)d_in[0];   // (4, 320, 9, 32, 32)
  const float* w0 = (const float*)d_in[1];   // (160, 320)
  const float* w1 = (const float*)d_in[2];   // (160, 320)
  float* out = (float*)d_out;                // (4, 98, 8, 32, 32)

  float* f = (float*)d_ws;                   // [2][32][160][1024] = 40 MB

  // 1) projections: 64 (j,n) blocks x 10 row-tiles, 4 waves each
  proj_kernel<<<dim3(64, 10), 128, 0, stream>>>(x, w0, w1, f);
  // 2) fused corr + pool + sample: one block per (image row, batch)
  corr_fused_kernel<<<dim3(32, 32), 256, 0, stream>>>(f, out);
}